// mincutnet_72524817760838
// MI455X (gfx1250) — compile-verified
//
#include <hip/hip_runtime.h>
#include <hip/hip_bf16.h>

// ---- fixed problem shapes (reference constants) ----
#define Bg    32
#define Nn    2000
#define Kc    20
#define EMBc  16
#define HIDc  16
#define INC   128
#define OUTC  10
#define DEGc  32
#define NTOT  (Bg * Nn)        // 64000
#define ETOT  (NTOT * DEGc)    // 2,048,000
#define ESL   (ETOT + NTOT)    // 2,112,000

typedef float v2f __attribute__((ext_vector_type(2)));
typedef float v8f __attribute__((ext_vector_type(8)));

__device__ __forceinline__ float lrelu(float v) { return v > 0.f ? v : 0.2f * v; }

__device__ __forceinline__ void atomicMaxF(float* addr, float val) {
    if (val >= 0.f) atomicMax((int*)addr, __float_as_int(val));
    else            atomicMin((unsigned int*)addr, __float_as_uint(val));
}

// ------------------------------------------------------------------
// WMMA f32 GEMM:  out[M x 16] = A[M x Kdim] * W[Kdim x 16]
// One wave per 16-row tile; V_WMMA_F32_16X16X4_F32, K stepped by 4.
// A layout (16x4 f32): lane L (0-15): M=L, K={0,1}; lane L (16-31): M=L-16, K={2,3}
// B layout (4x16 f32): symmetric: lane L holds N=L%16, K={0,1} or {2,3}
// C/D: vgpr v, lanes 0-15 -> M=v, lanes 16-31 -> M=v+8; N=lane%16
// ------------------------------------------------------------------
__global__ __launch_bounds__(256) void k_wmma_gemm_n16(
    const float* __restrict__ A, const float* __restrict__ W,
    float* __restrict__ out, int M, int Kdim) {
    int wave = threadIdx.x >> 5;
    int lane = threadIdx.x & 31;
    int tile = blockIdx.x * 8 + wave;            // 16 rows per tile
    if (tile * 16 >= M) return;                  // whole-wave uniform branch
    int rowN = lane & 15;                        // M (for A) / N (for B, C/D)
    int kgrp = (lane >> 4) << 1;                 // 0 or 2
    const float* arow = A + (size_t)(tile * 16 + rowN) * Kdim;
    v8f c = {};
    for (int k0 = 0; k0 < Kdim; k0 += 4) {
        v2f a, b;
        a.x = arow[k0 + kgrp];
        a.y = arow[k0 + kgrp + 1];
        b.x = W[(size_t)(k0 + kgrp) * 16 + rowN];
        b.y = W[(size_t)(k0 + kgrp + 1) * 16 + rowN];
        c = __builtin_amdgcn_wmma_f32_16x16x4_f32(false, a, false, b,
                                                  (short)0, c, false, false);
    }
    int mhalf = (lane >> 4) * 8;
#pragma unroll
    for (int v = 0; v < 8; ++v)
        out[(size_t)(tile * 16 + mhalf + v) * 16 + rowN] = c[v];
}

// ---------------- GCN ----------------
__global__ void k_deg_init(float* deg) {
    int i = blockIdx.x * blockDim.x + threadIdx.x;
    if (i < NTOT) deg[i] = 1.0f;                 // self loop
}
__global__ void k_deg_edges(const int* __restrict__ col, float* deg) {
    int e = blockIdx.x * blockDim.x + threadIdx.x;
    if (e < ETOT) atomicAdd(&deg[col[e]], 1.0f);
}
__global__ void k_dinv(float* deg) {
    int i = blockIdx.x * blockDim.x + threadIdx.x;
    if (i < NTOT) deg[i] = rsqrtf(deg[i]);
}
__global__ void k_gcn_self(const float* __restrict__ dinv,
                           const float* __restrict__ xw, float* h) {
    int idx = blockIdx.x * blockDim.x + threadIdx.x;
    if (idx >= NTOT * 16) return;
    int i = idx >> 4;
    float d = dinv[i];
    h[idx] = d * d * xw[idx];
}
__global__ void k_gcn_edges(const int* __restrict__ row, const int* __restrict__ col,
                            const float* __restrict__ dinv,
                            const float* __restrict__ xw, float* h) {
    int idx = blockIdx.x * blockDim.x + threadIdx.x;
    if (idx >= ETOT * 16) return;
    int e = idx >> 4, f = idx & 15;
    int r = row[e], c = col[e];
    float nrm = dinv[r] * dinv[c];
    atomicAdd(&h[(size_t)c * 16 + f], nrm * xw[(size_t)r * 16 + f]);
}
__global__ void k_gcn_final(float* h, const float* __restrict__ gcn_b) {
    int idx = blockIdx.x * blockDim.x + threadIdx.x;
    if (idx >= NTOT * 16) return;
    h[idx] = fmaxf(h[idx] + gcn_b[idx & 15], 0.f);
}

// ---------------- GAT ----------------
__global__ void k_att_dots(const float* __restrict__ hw,
                           const float* __restrict__ a_src,
                           const float* __restrict__ a_dst,
                           float* asrc, float* adst, float* m) {
    int i = blockIdx.x * blockDim.x + threadIdx.x;
    if (i >= NTOT) return;
    float s = 0.f, d = 0.f;
#pragma unroll
    for (int f = 0; f < 16; ++f) {
        float v = hw[(size_t)i * 16 + f];
        s += v * a_src[f];
        d += v * a_dst[f];
    }
    asrc[i] = s; adst[i] = d;
    m[i] = lrelu(s + d);                         // self-loop seed for segment max
}
__global__ void k_att_max_edges(const int* __restrict__ row, const int* __restrict__ col,
                                const float* __restrict__ asrc,
                                const float* __restrict__ adst, float* m) {
    int e = blockIdx.x * blockDim.x + threadIdx.x;
    if (e >= ETOT) return;
    int r = row[e], c = col[e];
    atomicMaxF(&m[c], lrelu(asrc[r] + adst[c]));
}
__global__ void k_att_self_exp(const float* __restrict__ asrc, const float* __restrict__ adst,
                               const float* __restrict__ m, float* denom, float* alpha) {
    int i = blockIdx.x * blockDim.x + threadIdx.x;
    if (i >= NTOT) return;
    float ee = __expf(lrelu(asrc[i] + adst[i]) - m[i]);
    denom[i] = ee;
    alpha[ETOT + i] = ee;                        // self loops follow edges
}
__global__ void k_att_edge_exp(const int* __restrict__ row, const int* __restrict__ col,
                               const float* __restrict__ asrc, const float* __restrict__ adst,
                               const float* __restrict__ m, float* denom, float* alpha) {
    int e = blockIdx.x * blockDim.x + threadIdx.x;
    if (e >= ETOT) return;
    int r = row[e], c = col[e];
    float ee = __expf(lrelu(asrc[r] + adst[c]) - m[c]);
    alpha[e] = ee;
    atomicAdd(&denom[c], ee);
}
__global__ void k_att_norm(const int* __restrict__ col, const float* __restrict__ denom,
                           float* alpha) {
    int idx = blockIdx.x * blockDim.x + threadIdx.x;
    if (idx >= ESL) return;
    int c = (idx < ETOT) ? col[idx] : (idx - ETOT);
    alpha[idx] /= denom[c];
}
__global__ void k_gat_self(const float* __restrict__ alpha, const float* __restrict__ hw,
                           float* nemb) {
    int idx = blockIdx.x * blockDim.x + threadIdx.x;
    if (idx >= NTOT * 16) return;
    nemb[idx] = alpha[ETOT + (idx >> 4)] * hw[idx];
}
__global__ void k_gat_edges(const int* __restrict__ row, const int* __restrict__ col,
                            const float* __restrict__ alpha, const float* __restrict__ hw,
                            float* nemb) {
    int idx = blockIdx.x * blockDim.x + threadIdx.x;
    if (idx >= ETOT * 16) return;
    int e = idx >> 4, f = idx & 15;
    int r = row[e], c = col[e];
    atomicAdd(&nemb[(size_t)c * 16 + f], alpha[e] * hw[(size_t)r * 16 + f]);
}
__global__ void k_gat_final(float* nemb, const float* __restrict__ gat_b, float* xd) {
    int idx = blockIdx.x * blockDim.x + threadIdx.x;
    if (idx >= NTOT * 16) return;
    float v = nemb[idx] + gat_b[idx & 15];
    nemb[idx] = v;                               // node_emb output
    xd[idx] = fmaxf(v, 0.f);
}

// ---------------- pool assignment softmax: s = softmax(xd @ pool_w + pool_b) ----------------
__global__ void k_softmax_s(const float* __restrict__ xd, const float* __restrict__ pool_w,
                            const float* __restrict__ pool_b, float* s) {
    int i = blockIdx.x * blockDim.x + threadIdx.x;
    if (i >= NTOT) return;
    float lg[Kc];
#pragma unroll
    for (int k = 0; k < Kc; ++k) lg[k] = pool_b[k];
#pragma unroll
    for (int f = 0; f < 16; ++f) {
        float xv = xd[(size_t)i * 16 + f];
#pragma unroll
        for (int k = 0; k < Kc; ++k) lg[k] += xv * pool_w[f * Kc + k];
    }
    float mx = lg[0];
#pragma unroll
    for (int k = 1; k < Kc; ++k) mx = fmaxf(mx, lg[k]);
    float sum = 0.f;
#pragma unroll
    for (int k = 0; k < Kc; ++k) { lg[k] = __expf(lg[k] - mx); sum += lg[k]; }
    float inv = 1.f / sum;
#pragma unroll
    for (int k = 0; k < Kc; ++k) s[(size_t)i * Kc + k] = lg[k] * inv;
}

// ---------------- out_x[b,k,f] = sum_n s[b,n,k] * xd[b,n,f] ----------------
__global__ __launch_bounds__(320) void k_outx(const float* __restrict__ s,
                                              const float* __restrict__ xd,
                                              float* out_x) {
    int b = blockIdx.x, t = threadIdx.x;         // t in [0,320): (k,f)
    int k = t / 16, f = t & 15;
    float acc = 0.f;
    const float* sb = s + (size_t)b * Nn * Kc;
    const float* xb = xd + (size_t)b * Nn * 16;
    for (int n = 0; n < Nn; ++n)
        acc += sb[(size_t)n * Kc + k] * xb[(size_t)n * 16 + f];
    out_x[(size_t)b * (Kc * 16) + t] = acc;
}

// ---------------- out_adj[b] = sum_edges outer(s[row], s[col])  (LDS accumulation) ----------------
#define ADJ_CHUNKS 16
__global__ __launch_bounds__(256) void k_zero_adj(float* out_adj) {
    int idx = blockIdx.x * blockDim.x + threadIdx.x;
    if (idx < Bg * Kc * Kc) out_adj[idx] = 0.f;
}
__global__ __launch_bounds__(256) void k_outadj(const int* __restrict__ row,
                                                const int* __restrict__ col,
                                                const float* __restrict__ s,
                                                float* out_adj) {
    __shared__ float acc[Kc * Kc];
    int b = blockIdx.x / ADJ_CHUNKS;
    int chunk = blockIdx.x % ADJ_CHUNKS;
    const int epg = Nn * DEGc;                   // 64000 edges per graph
    const int cpe = epg / ADJ_CHUNKS;            // 4000 per block
    for (int i = threadIdx.x; i < Kc * Kc; i += 256) acc[i] = 0.f;
    __syncthreads();
    int base = b * epg + chunk * cpe;
    for (int le = threadIdx.x; le < cpe; le += 256) {
        int e = base + le;
        int i = row[e] % Nn, j = col[e] % Nn;
        const float* sr = s + (size_t)(b * Nn + i) * Kc;
        const float* sc = s + (size_t)(b * Nn + j) * Kc;
        float srr[Kc], scc[Kc];
#pragma unroll
        for (int k = 0; k < Kc; ++k) { srr[k] = sr[k]; scc[k] = sc[k]; }
#pragma unroll
        for (int k = 0; k < Kc; ++k) {
            float rv = srr[k];
#pragma unroll
            for (int l = 0; l < Kc; ++l)
                atomicAdd(&acc[k * Kc + l], rv * scc[l]);
        }
    }
    __syncthreads();
    for (int i = threadIdx.x; i < Kc * Kc; i += 256)
        atomicAdd(&out_adj[(size_t)b * Kc * Kc + i], acc[i]);
}

// ---------------- zero diagonal + symmetric degree normalization ----------------
__global__ __launch_bounds__(400) void k_adj_norm(float* out_adj) {
    __shared__ float a[Kc * Kc];
    __shared__ float dsq[Kc];
    int b = blockIdx.x, t = threadIdx.x;         // t in [0,400)
    int k = t / Kc, l = t % Kc;
    float v = out_adj[(size_t)b * Kc * Kc + t];
    if (k == l) v = 0.f;
    a[t] = v;
    __syncthreads();
    if (t < Kc) {
        float sum = 0.f;
#pragma unroll
        for (int j = 0; j < Kc; ++j) sum += a[t * Kc + j];
        dsq[t] = sqrtf(sum) + 1e-15f;
    }
    __syncthreads();
    out_adj[(size_t)b * Kc * Kc + t] = a[t] / (dsq[k] * dsq[l]);
}

// ---------------- DenseGraphConv + mean pool + linear head ----------------
__global__ __launch_bounds__(320) void k_final(const float* __restrict__ out_adj,
                                               const float* __restrict__ out_x,
                                               const float* __restrict__ w_rel,
                                               const float* __restrict__ b_rel,
                                               const float* __restrict__ w_root,
                                               const float* __restrict__ lin2_w,
                                               const float* __restrict__ lin2_b,
                                               float* o_out, float* o_gemb) {
    __shared__ float adjS[Kc * Kc];
    __shared__ float oxS[Kc * 16];
    __shared__ float axS[Kc * 16];
    __shared__ float xcS[Kc * 16];
    __shared__ float gembS[16];
    int b = blockIdx.x, t = threadIdx.x;         // t in [0,320): (k,f)
    for (int i = t; i < Kc * Kc; i += 320) adjS[i] = out_adj[(size_t)b * Kc * Kc + i];
    oxS[t] = out_x[(size_t)b * Kc * 16 + t];
    __syncthreads();
    int k = t / 16, f = t & 15;
    float ax = 0.f;
#pragma unroll
    for (int l = 0; l < Kc; ++l) ax += adjS[k * Kc + l] * oxS[l * 16 + f];
    axS[t] = ax;
    __syncthreads();
    float xc = b_rel[f];
#pragma unroll
    for (int g = 0; g < 16; ++g)
        xc += axS[k * 16 + g] * w_rel[g * 16 + f] + oxS[k * 16 + g] * w_root[g * 16 + f];
    xcS[t] = xc;
    __syncthreads();
    if (t < 16) {
        float gm = 0.f;
#pragma unroll
        for (int kk = 0; kk < Kc; ++kk) gm += xcS[kk * 16 + t];
        gm *= (1.f / (float)Kc);
        gembS[t] = gm;
        o_gemb[(size_t)b * 16 + t] = gm;
    }
    __syncthreads();
    if (t < OUTC) {
        float o = lin2_b[t];
#pragma unroll
        for (int ff = 0; ff < 16; ++ff) o += gembS[ff] * lin2_w[ff * OUTC + t];
        o_out[(size_t)b * OUTC + t] = o;
    }
}

extern "C" void kernel_launch(void* const* d_in, const int* in_sizes, int n_in,
                              void* d_out, int out_size, void* d_ws, size_t ws_size,
                              hipStream_t stream) {
    const float* x       = (const float*)d_in[0];
    const int*   ei      = (const int*)d_in[1];
    const float* gcn_w   = (const float*)d_in[3];
    const float* gcn_b   = (const float*)d_in[4];
    const float* gat_w   = (const float*)d_in[5];
    const float* a_src   = (const float*)d_in[6];
    const float* a_dst   = (const float*)d_in[7];
    const float* gat_b   = (const float*)d_in[8];
    const float* pool_w  = (const float*)d_in[9];
    const float* pool_b  = (const float*)d_in[10];
    const float* w_rel   = (const float*)d_in[11];
    const float* b_rel   = (const float*)d_in[12];
    const float* w_root  = (const float*)d_in[13];
    const float* lin2_w  = (const float*)d_in[14];
    const float* lin2_b  = (const float*)d_in[15];

    const int* row = ei;
    const int* col = ei + ETOT;

    // workspace partition (floats)
    float* wf    = (float*)d_ws;
    float* xw    = wf;                           // NTOT*16
    float* h     = wf + 1024000;                 // NTOT*16
    float* hw    = wf + 2048000;                 // NTOT*16
    float* xd    = wf + 3072000;                 // NTOT*16
    float* dinv  = wf + 4096000;                 // NTOT (deg -> dinv in place)
    float* asrc  = wf + 4160000;                 // NTOT
    float* adst  = wf + 4224000;                 // NTOT
    float* mbuf  = wf + 4288000;                 // NTOT
    float* denom = wf + 4352000;                 // NTOT
    float* outx  = wf + 4416000;                 // Bg*Kc*16
    float* outadj= wf + 4426240;                 // Bg*Kc*Kc

    // output partition (return order: out, g_emb, node_emb, alpha, s)
    float* outp   = (float*)d_out;
    float* o_out  = outp;                        // 320
    float* o_gemb = outp + 320;                  // 512
    float* o_nemb = outp + 832;                  // NTOT*16
    float* o_alpha= outp + 1024832;              // ESL
    float* o_s    = outp + 3136832;              // NTOT*Kc

    const int TB = 256;
    const int gNode   = (NTOT + TB - 1) / TB;        // 250
    const int gNode16 = (NTOT * 16 + TB - 1) / TB;   // 4000
    const int gEdge   = (ETOT + TB - 1) / TB;        // 8000
    const int gEdge16 = (ETOT * 16 + TB - 1) / TB;   // 128000
    const int gESL    = (ESL + TB - 1) / TB;         // 8250
    const int gGemm   = (NTOT / 16) / 8;             // 500 blocks x 8 waves, exact

    // --- GCN: D^-1/2 (A+I) D^-1/2 X W, relu ---
    k_deg_init<<<gNode, TB, 0, stream>>>(dinv);
    k_deg_edges<<<gEdge, TB, 0, stream>>>(col, dinv);
    k_dinv<<<gNode, TB, 0, stream>>>(dinv);
    k_wmma_gemm_n16<<<gGemm, TB, 0, stream>>>(x, gcn_w, xw, NTOT, INC);
    k_gcn_self<<<gNode16, TB, 0, stream>>>(dinv, xw, h);
    k_gcn_edges<<<gEdge16, TB, 0, stream>>>(row, col, dinv, xw, h);
    k_gcn_final<<<gNode16, TB, 0, stream>>>(h, gcn_b);

    // --- GAT: hw = h @ gat_w, segment softmax over dst ---
    k_wmma_gemm_n16<<<gGemm, TB, 0, stream>>>(h, gat_w, hw, NTOT, HIDc);
    k_att_dots<<<gNode, TB, 0, stream>>>(hw, a_src, a_dst, asrc, adst, mbuf);
    k_att_max_edges<<<gEdge, TB, 0, stream>>>(row, col, asrc, adst, mbuf);
    k_att_self_exp<<<gNode, TB, 0, stream>>>(asrc, adst, mbuf, denom, o_alpha);
    k_att_edge_exp<<<gEdge, TB, 0, stream>>>(row, col, asrc, adst, mbuf, denom, o_alpha);
    k_att_norm<<<gESL, TB, 0, stream>>>(col, denom, o_alpha);
    k_gat_self<<<gNode16, TB, 0, stream>>>(o_alpha, hw, o_nemb);
    k_gat_edges<<<gEdge16, TB, 0, stream>>>(row, col, o_alpha, hw, o_nemb);
    k_gat_final<<<gNode16, TB, 0, stream>>>(o_nemb, gat_b, xd);

    // --- mincut pool (only s / out_x / out_adj feed the output) ---
    k_softmax_s<<<gNode, TB, 0, stream>>>(xd, pool_w, pool_b, o_s);
    k_outx<<<Bg, 320, 0, stream>>>(o_s, xd, outx);
    k_zero_adj<<<(Bg * Kc * Kc + TB - 1) / TB, TB, 0, stream>>>(outadj);
    k_outadj<<<Bg * ADJ_CHUNKS, TB, 0, stream>>>(row, col, o_s, outadj);
    k_adj_norm<<<Bg, 400, 0, stream>>>(outadj);

    // --- DenseGraphConv + head ---
    k_final<<<Bg, 320, 0, stream>>>(outadj, outx, w_rel, b_rel, w_root,
                                    lin2_w, lin2_b, o_out, o_gemb);
}